// Net_68453188764069
// MI455X (gfx1250) — compile-verified
//
#include <hip/hip_runtime.h>

typedef __attribute__((ext_vector_type(2))) float    v2f;
typedef __attribute__((ext_vector_type(8))) float    v8f;
typedef __attribute__((ext_vector_type(4))) unsigned v4u;
typedef __attribute__((ext_vector_type(4))) int      v4i;
typedef __attribute__((ext_vector_type(8))) int      v8i;

#define B_    16
#define N_    64
#define S_    4
#define F0_   8
#define C_    32
#define GSTR  208   // row stride (floats) for G and Mt
#define KPAD1 64    // layer1: 6*8+1 = 49 -> pad 64
#define KPAD2 208   // layer2: 6*32+1 = 193 -> pad 208

// LDS float offsets
#define OFF_A     0        // a   [64][64]           4096
#define OFF_H1    4096     // h1  [64][32]           2048
#define OFF_H2    6144     // h2  [64][32]           2048
#define OFF_MASK  8192     // mask[64]               64
#define OFF_RED   8256     // red [64]               64
#define OFF_G     8320     // G   [64][208]          13312
#define OFF_M     21632    // Mt  [32][208]          6656
#define OFF_E     OFF_G    // raw e [64*64*4] OVERLAYS G+M (dead before G written)
#define OFF_AE    28288    // ae  [64][5][64]        20480
#define SM_FLOATS 48768    // 195072 bytes (< 320KB/WGP)

#if defined(__has_builtin)
# if __has_builtin(__builtin_amdgcn_tensor_load_to_lds) && __has_builtin(__builtin_amdgcn_s_wait_tensorcnt)
#  define USE_TDM 1
# endif
#endif

#if defined(USE_TDM)
# if __has_include(<hip/amd_detail/amd_gfx1250_TDM.h>)
   // amdgpu-toolchain (clang-23): 6-arg form
#  define TDM_LOAD(g0, g1, g2, g3) \
     __builtin_amdgcn_tensor_load_to_lds((g0), (g1), (g2), (g3), (v8i)(0), 0)
# else
   // ROCm 7.2 (clang-22): 5-arg form
#  define TDM_LOAD(g0, g1, g2, g3) \
     __builtin_amdgcn_tensor_load_to_lds((g0), (g1), (g2), (g3), 0)
# endif
#endif

// One ECC layer. AE[t][s][i] (s<4: a*e channel s; s==4: a) is prebuilt in LDS.
//   G[t, s*Fin+y] = sum_i AE[t,s,i] * hin[i,y]        (WMMA, step A)
//   G[t, 4*Fin+y] = sum_i AE[t,4,i] * hin[i,y]        (xa, same WMMA form)
//   G[t, 5*Fin+y] = hin[t,y]   (root term),  G[t,6*Fin] = 1 (bias column)
//   Mt[x][k] stacks wk / bk / wr / bias;  hout = relu(mask .* (G @ M))  (WMMA, step B)
__device__ __forceinline__ void ecc_layer(
    const float* ae, const float* hin, float* hout, const float* mask,
    float* G, float* Mt,
    const float* __restrict__ wk, const float* __restrict__ bk,
    const float* __restrict__ wr, const float* __restrict__ bias,
    const int Fin, const int Kpad)
{
  const int tid  = (int)threadIdx.x;
  const int lane = tid & 31;
  const int wave = tid >> 5;     // 0..7
  const int half = lane >> 4;    // 0 or 1
  const int lm   = lane & 15;

  // ---- Build Mt (32 x Kpad, transposed) and pre-fill G (64 x Kpad) ----
  {
    const int x = tid & 31;
    for (int k = tid >> 5; k < Kpad; k += 8) {
      float v = 0.f;
      if      (k < 4 * Fin)  v = wk[(k / Fin) * (C_ * Fin) + x * Fin + (k % Fin)];
      else if (k < 5 * Fin)  v = bk[x * Fin + (k - 4 * Fin)];
      else if (k < 6 * Fin)  v = wr[(k - 5 * Fin) * C_ + x];
      else if (k == 6 * Fin) v = bias[x];
      Mt[x * GSTR + k] = v;
    }
    const int t = tid & 63;
    for (int k = tid >> 6; k < Kpad; k += 4) {
      float v = 0.f;
      if (k >= 5 * Fin && k < 6 * Fin) v = hin[t * C_ + (k - 5 * Fin)];
      else if (k == 6 * Fin)           v = 1.0f;
      G[t * GSTR + k] = v;
    }
  }
  __syncthreads();

  // ---- Step A: V_WMMA_F32_16X16X4_F32 over K=64 (sum over source nodes) ----
  const int ntiles = (Fin + 15) >> 4;        // 1 (layer1) or 2 (layer2)
  const int jobsG  = 16 * ntiles;            // 4 targets x 4 edge-chans per tile
  const int jobs   = jobsG + 4 * ntiles;     // + xa tiles: 16 targets each
  for (int job = wave; job < jobs; job += 8) {
    const bool isG = (job < jobsG);
    const int  jj  = isG ? job : (job - jobsG);
    const int  tg  = (ntiles == 2) ? (jj >> 1) : jj;
    const int  nt  = (ntiles == 2) ? (jj & 1) : 0;
    const int  t_a = isG ? (tg * 4 + (lm >> 2)) : (tg * 16 + lm);
    const int  s_a = isG ? (lm & 3) : 4;
    const float* arow = ae + (t_a * 5 + s_a) * N_;   // contiguous in i
    const float* bcol = hin + nt * 16 + lm;
    v8f acc = {};
    for (int k0 = 0; k0 < N_; k0 += 4) {
      const int i0 = k0 + 2 * half;
      const v2f af = *(const v2f*)(arow + i0);       // ds_load_b64
      v2f bf;
      bf.x = bcol[i0 * C_];
      bf.y = bcol[(i0 + 1) * C_];
      acc = __builtin_amdgcn_wmma_f32_16x16x4_f32(
          false, af, false, bf, (short)0, acc, false, false);
    }
    const int y = nt * 16 + lm;
    if (y < Fin) {
      for (int r = 0; r < 8; ++r) {
        const int m   = r + 8 * half;                // D row
        const int t   = isG ? (tg * 4 + (m >> 2)) : (tg * 16 + m);
        const int col = isG ? ((m & 3) * Fin + y) : (4 * Fin + y);
        G[t * GSTR + col] = acc[r];
      }
    }
  }
  __syncthreads();

  // ---- Step B: hout = relu(mask .* (G @ M)) ----
  {
    const int ttile = wave >> 1;   // 0..3
    const int xtile = wave & 1;    // 0..1
    const float* ga = G  + (ttile * 16 + lm) * GSTR;
    const float* mb = Mt + (xtile * 16 + lm) * GSTR;
    v8f acc = {};
    for (int k0 = 0; k0 < Kpad; k0 += 4) {
      const int kk = k0 + 2 * half;
      const v2f af = *(const v2f*)(ga + kk);         // ds_load_b64
      const v2f bf = *(const v2f*)(mb + kk);         // ds_load_b64
      acc = __builtin_amdgcn_wmma_f32_16x16x4_f32(
          false, af, false, bf, (short)0, acc, false, false);
    }
    for (int r = 0; r < 8; ++r) {
      const int t = ttile * 16 + r + 8 * half;
      const int x = xtile * 16 + lm;
      float v = acc[r] * mask[t];
      hout[t * C_ + x] = v > 0.f ? v : 0.f;
    }
  }
  __syncthreads();
}

__global__ void __launch_bounds__(256)
ecc_net_kernel(const float* __restrict__ x,  const float* __restrict__ a,
               const float* __restrict__ e,
               const float* __restrict__ w1k, const float* __restrict__ b1k,
               const float* __restrict__ w1r, const float* __restrict__ b1,
               const float* __restrict__ w2k, const float* __restrict__ b2k,
               const float* __restrict__ w2r, const float* __restrict__ b2,
               const float* __restrict__ wd,  const float* __restrict__ bd,
               float* __restrict__ out)
{
  extern __shared__ float sm[];
  float* a_s  = sm + OFF_A;
  float* h1   = sm + OFF_H1;
  float* h2   = sm + OFF_H2;
  float* mask = sm + OFF_MASK;
  float* red  = sm + OFF_RED;
  float* G    = sm + OFF_G;
  float* Mt   = sm + OFF_M;
  float* E    = sm + OFF_E;    // raw e slab (overlays G/Mt; dead before they're written)
  float* ae   = sm + OFF_AE;

  const int b    = (int)blockIdx.x;
  const int tid  = (int)threadIdx.x;
  const int wave = tid >> 5;
  const float* xg = x + (size_t)b * N_ * (F0_ + 1);
  const float* ag = a + (size_t)b * N_ * N_;
  const float* eg = e + (size_t)b * N_ * N_ * S_;

#if defined(USE_TDM)
  // Kick off the TDM DMA of this batch's 64KB e-slab into LDS first, so it
  // overlaps with staging a/x below. One issue per block (wave 0).
  if (wave == 0) {
    const unsigned long long ga64 = (unsigned long long)(uintptr_t)eg;
    v4u g0;
    g0.x = 1u;                                            // count=1 (valid D#)
    g0.y = (unsigned)(OFF_E * 4);                         // lds_addr (bytes)
    g0.z = (unsigned)(ga64 & 0xFFFFFFFFull);              // global_addr[31:0]
    g0.w = (unsigned)((ga64 >> 32) & 0x1FFFFFFull)        // global_addr[56:32]
         | (2u << 30);                                    // type=2 ("image")
    v8i g1;
    g1[0] = (int)(2u << 16);                              // data_size=2 (4B)
    g1[1] = (int)(0x4000u << 16);                         // tensor_dim0[15:0]=16384
    g1[2] = (int)((0x4000u >> 16) | (1u << 16));          // dim0[31:16], dim1[15:0]=1
    g1[3] = (int)(0x4000u << 16);                         // dim1[31:16]=0, tile_dim0=16384
    g1[4] = 1;                                            // tile_dim1=1, tile_dim2=0
    g1[5] = 0x4000;                                       // tensor_dim0_stride=16384
    g1[6] = 0;
    g1[7] = 0;
    const v4i g2 = (v4i)(0);
    const v4i g3 = (v4i)(0);
    TDM_LOAD(g0, g1, g2, g3);
  }
#else
  // Fallback: plain vectorized copy of the e-slab into LDS.
  for (int idx = tid; idx < (N_ * N_ * S_) / 4; idx += 256)
    ((float4*)E)[idx] = ((const float4*)eg)[idx];
#endif

  // Stage adjacency, node features (zero-padded to 32 cols), mask.
  for (int idx = tid; idx < N_ * N_; idx += 256) a_s[idx] = ag[idx];
  for (int idx = tid; idx < N_ * C_; idx += 256) {
    const int t = idx >> 5, yy = idx & 31;
    h1[idx] = (yy < F0_) ? xg[t * (F0_ + 1) + yy] : 0.f;
  }
  for (int t = tid; t < N_; t += 256) mask[t] = xg[t * (F0_ + 1) + F0_];
  __syncthreads();

#if defined(USE_TDM)
  if (wave == 0) __builtin_amdgcn_s_wait_tensorcnt(0);    // s_wait_tensorcnt 0
  __syncthreads();
#endif

  // Build AE[t][s][i]: s<4 -> a[t,i]*e[t,i,s]; s==4 -> a[t,i] (xa channel).
  // Contiguous in i so step-A A-fragments are single ds_load_b64's.
#pragma unroll
  for (int s = 0; s < 5; ++s) {
    for (int idx = tid; idx < N_ * N_; idx += 256) {
      const int t = idx >> 6, i = idx & 63;
      const float av = a_s[idx];
      const float ev = (s < 4) ? E[idx * S_ + s] : 1.0f;
      ae[(t * 5 + s) * N_ + i] = av * ev;
    }
  }
  __syncthreads();   // E region is now dead; G/Mt may overwrite it

  ecc_layer(ae, h1, h2, mask, G, Mt, w1k, b1k, w1r, b1, F0_, KPAD1);
  ecc_layer(ae, h2, h1, mask, G, Mt, w2k, b2k, w2r, b2, C_,  KPAD2);

  // pooled = sum_t mask[t]*h[t,:];  out[b] = pooled @ wd + bd   (NOUT=1)
  if (tid < N_) {
    float v = 0.f;
    for (int xx = 0; xx < C_; ++xx) v += h1[tid * C_ + xx] * wd[xx];
    red[tid] = v * mask[tid];
  }
  __syncthreads();
  for (int off = 32; off > 0; off >>= 1) {
    if (tid < off && tid + off < N_) red[tid] += red[tid + off];
    __syncthreads();
  }
  if (tid == 0) out[b] = red[0] + bd[0];
}

extern "C" void kernel_launch(void* const* d_in, const int* in_sizes, int n_in,
                              void* d_out, int out_size, void* d_ws, size_t ws_size,
                              hipStream_t stream) {
  (void)in_sizes; (void)n_in; (void)out_size; (void)d_ws; (void)ws_size;
  const float* x   = (const float*)d_in[0];
  const float* a   = (const float*)d_in[1];
  const float* e   = (const float*)d_in[2];
  const float* w1k = (const float*)d_in[3];
  const float* b1k = (const float*)d_in[4];
  const float* w1r = (const float*)d_in[5];
  const float* b1  = (const float*)d_in[6];
  const float* w2k = (const float*)d_in[7];
  const float* b2k = (const float*)d_in[8];
  const float* w2r = (const float*)d_in[9];
  const float* b2  = (const float*)d_in[10];
  const float* wd  = (const float*)d_in[11];
  const float* bd  = (const float*)d_in[12];
  const size_t shbytes = (size_t)SM_FLOATS * sizeof(float);
  ecc_net_kernel<<<dim3(B_), dim3(256), shbytes, stream>>>(
      x, a, e, w1k, b1k, w1r, b1, w2k, b2k, w2r, b2, wd, bd, (float*)d_out);
}